// MessagePassing_10651518894406
// MI455X (gfx1250) — compile-verified
//
#include <hip/hip_runtime.h>
#include <stdint.h>

#define NUM_NODES 10000
#define NUM_EDGES 640000
#define D_FEAT    128
#define D4        (D_FEAT / 4)          // 32 float4 per row -> one per lane
#define WAVES_PER_BLOCK 8               // 256 threads = 8 wave32

typedef float v4f __attribute__((ext_vector_type(4)));

// --- CDNA5 async global->LDS path (guarded; falls back to wave shuffles) ---
#if defined(__HIP_DEVICE_COMPILE__) && defined(__gfx1250__) && \
    __has_builtin(__builtin_amdgcn_global_load_async_to_lds_b32)
#define USE_ASYNC_LDS 1
typedef __attribute__((address_space(1))) int as1_int;
typedef __attribute__((address_space(3))) int as3_int;

__device__ __forceinline__ void async_id_load(const int* gsrc, int* lds_slot) {
    // AS1 pointer via inttoptr (64-bit global address).
    as1_int* gp = (as1_int*)(uintptr_t)(const void*)gsrc;
    // AS3 pointer: low 32 bits of a generic LDS pointer == LDS offset.
    as3_int* lp = (as3_int*)(uintptr_t)(uint32_t)(uintptr_t)lds_slot;
    __builtin_amdgcn_global_load_async_to_lds_b32(gp, lp, /*imm offset*/ 0,
                                                  /*cpol*/ 0);
}
__device__ __forceinline__ void wait_async(int n) {
#if __has_builtin(__builtin_amdgcn_s_wait_asynccnt)
    if (n == 0) __builtin_amdgcn_s_wait_asynccnt(0);
    else        __builtin_amdgcn_s_wait_asynccnt(1);
#else
    if (n == 0) asm volatile("s_wait_asynccnt 0" ::: "memory");
    else        asm volatile("s_wait_asynccnt 1" ::: "memory");
#endif
}
#else
#define USE_ASYNC_LDS 0
#endif

// ---------------------------------------------------------------------------
// Phase 4 (placed first in .text so the disasm snippet shows it):
// atomic-free aggregation. One wave32 per node; lane l owns float4 column l of
// the 128-wide row (32 lanes * 16B = 512B row, fully coalesced GLOBAL_LOAD_B128
// gathers out of L2-resident x). Edge-id chunks of 32 are double-buffer staged
// into LDS with CDNA5 async global->LDS loads: issue chunk c+1, then
// s_wait_asynccnt 1 (in-order completion) guarantees chunk c has landed, so the
// id fetch overlaps the current chunk's row gathers.
__global__ __launch_bounds__(WAVES_PER_BLOCK * 32)
void aggregate_kernel(const float* __restrict__ x,
                      const int*   __restrict__ ssrc,
                      const int*   __restrict__ offsets,
                      float*       __restrict__ out) {
    const int lane = threadIdx.x & 31;
    const int w    = threadIdx.x >> 5;
    const int node = blockIdx.x * WAVES_PER_BLOCK + w;
#if USE_ASYNC_LDS
    __shared__ int ids[WAVES_PER_BLOCK][2][32];   // wave-private double buffer
#endif
    if (node >= NUM_NODES) return;

    const v4f* __restrict__ x4 = (const v4f*)x;
    const int beg = offsets[node];
    const int end = offsets[node + 1];
    v4f acc = (v4f){0.f, 0.f, 0.f, 0.f};

#if USE_ASYNC_LDS
    const int nchunk = (end - beg + 31) >> 5;
    if (nchunk > 0) {
        {   // preload chunk 0 into slot 0
            const int cnt0 = (end - beg < 32) ? (end - beg) : 32;
            if (lane < cnt0) async_id_load(ssrc + beg + lane, &ids[w][0][lane]);
        }
        for (int c = 0; c < nchunk; ++c) {
            const int base = beg + (c << 5);
            int cnt = end - base;
            if (cnt > 32) cnt = 32;
            if (c + 1 < nchunk) {
                const int nbase = base + 32;
                int ncnt = end - nbase;
                if (ncnt > 32) ncnt = 32;
                if (nbase + 32 < end)          // prefetch chunk c+2 ids
                    __builtin_prefetch(&ssrc[nbase + 32], 0, 3);
                if (lane < ncnt)
                    async_id_load(ssrc + nbase + lane, &ids[w][(c + 1) & 1][lane]);
                wait_async(1);                 // chunk c complete (in-order)
            } else {
                wait_async(0);
            }
            const int* idp = ids[w][c & 1];
#pragma unroll 4
            for (int j = 0; j < cnt; ++j) {
                const int s = idp[j];
                const v4f v = x4[(size_t)s * D4 + lane];
                acc += v;
            }
        }
    }
#else
    for (int base = beg; base < end; base += 32) {
        int cnt = end - base;
        if (cnt > 32) cnt = 32;
        if (base + 32 < end)
            __builtin_prefetch(&ssrc[base + 32], 0, 3);
        const int sid = ssrc[base + ((lane < cnt) ? lane : 0)];
#pragma unroll 4
        for (int j = 0; j < cnt; ++j) {
            const int s = __shfl(sid, j, 32);  // wave32 ds_bpermute broadcast
            const v4f v = x4[(size_t)s * D4 + lane];
            acc += v;
        }
    }
#endif
    // Streaming write-once output: non-temporal B128 store.
    __builtin_nontemporal_store(acc, (v4f*)out + (size_t)node * D4 + lane);
}

// ---------------------------------------------------------------------------
// Phase 0: zero the degree histogram (d_ws is poisoned with 0xAA).
__global__ void zero_int_kernel(int* __restrict__ p, int n) {
    int i = blockIdx.x * blockDim.x + threadIdx.x;
    if (i < n) p[i] = 0;
}

// Phase 1: per-destination degree histogram. 640K int atomics (cheap vs 82M f32).
__global__ void degree_kernel(const int* __restrict__ dst, int* __restrict__ deg) {
    int e = blockIdx.x * blockDim.x + threadIdx.x;
    if (e < NUM_EDGES) atomicAdd(&deg[dst[e]], 1);
}

// Phase 2: exclusive prefix sum of 10000 degrees, single 1024-thread block
// (10 chunks x Hillis-Steele in LDS; trivially small). Also seeds the cursor.
#define SCAN_BLOCK 1024
__global__ void scan_kernel(const int* __restrict__ deg,
                            int* __restrict__ offsets,
                            int* __restrict__ cursor, int n) {
    __shared__ int buf[SCAN_BLOCK];
    __shared__ int carry;
    const int tid = threadIdx.x;
    if (tid == 0) carry = 0;
    __syncthreads();
    for (int base = 0; base < n; base += SCAN_BLOCK) {
        const int i = base + tid;
        const int v = (i < n) ? deg[i] : 0;
        buf[tid] = v;
        __syncthreads();
        // Hillis-Steele inclusive scan
        for (int off = 1; off < SCAN_BLOCK; off <<= 1) {
            int t = (tid >= off) ? buf[tid - off] : 0;
            __syncthreads();
            buf[tid] += t;
            __syncthreads();
        }
        const int incl = buf[tid];
        const int excl = carry + incl - v;
        if (i < n) { offsets[i] = excl; cursor[i] = excl; }
        __syncthreads();                       // everyone has read old carry
        if (tid == SCAN_BLOCK - 1) carry += incl;
        __syncthreads();
    }
    if (tid == 0) offsets[n] = carry;          // total = NUM_EDGES
}

// Phase 3: counting-sort edges by destination (CSR build).
__global__ void bin_kernel(const int* __restrict__ src, const int* __restrict__ dst,
                           int* __restrict__ cursor, int* __restrict__ ssrc) {
    int e = blockIdx.x * blockDim.x + threadIdx.x;
    if (e < NUM_EDGES) {
        int p = atomicAdd(&cursor[dst[e]], 1);
        ssrc[p] = src[e];
    }
}

// ---------------------------------------------------------------------------
extern "C" void kernel_launch(void* const* d_in, const int* in_sizes, int n_in,
                              void* d_out, int out_size, void* d_ws, size_t ws_size,
                              hipStream_t stream) {
    const float* x  = (const float*)d_in[0];
    const int*   ei = (const int*)d_in[1];    // edge_index flattened [2, E]
    const int*   src = ei;                    // row 0
    const int*   dst = ei + NUM_EDGES;        // row 1
    float*       out = (float*)d_out;

    // Workspace layout (ints), padded offsets for alignment: ~2.75 MB total
    int* deg     = (int*)d_ws;                // [10000]
    int* offsets = deg + 16384;               // [10001]
    int* cursor  = offsets + 16384;           // [10000]
    int* ssrc    = cursor + 16384;            // [640000]

    zero_int_kernel<<<(NUM_NODES + 255) / 256, 256, 0, stream>>>(deg, NUM_NODES);
    degree_kernel<<<(NUM_EDGES + 255) / 256, 256, 0, stream>>>(dst, deg);
    scan_kernel<<<1, SCAN_BLOCK, 0, stream>>>(deg, offsets, cursor, NUM_NODES);
    bin_kernel<<<(NUM_EDGES + 255) / 256, 256, 0, stream>>>(src, dst, cursor, ssrc);

    const int blocks = (NUM_NODES + WAVES_PER_BLOCK - 1) / WAVES_PER_BLOCK;
    aggregate_kernel<<<blocks, WAVES_PER_BLOCK * 32, 0, stream>>>(
        x, ssrc, offsets, out);
}